// APPNP_17884243821383
// MI455X (gfx1250) — compile-verified
//
#include <hip/hip_runtime.h>

#define ALPHA 0.1f
#define NITER 10

typedef __attribute__((ext_vector_type(2))) float v2f;
typedef __attribute__((ext_vector_type(8))) float v8f;

// ---------------------------------------------------------------------------
// GCN normalization: deg (with self loop) -> dinv -> effective edge weights
// ---------------------------------------------------------------------------
__global__ void __launch_bounds__(256) deg_init_kernel(float* __restrict__ deg, int N) {
  int i = blockIdx.x * 256 + threadIdx.x;
  if (i < N) deg[i] = 1.0f;  // self-loop contributes weight 1
}

__global__ void __launch_bounds__(256) deg_accum_kernel(const int* __restrict__ ei,
                                                        const float* __restrict__ ew,
                                                        float* __restrict__ deg, int E) {
  int e = blockIdx.x * 256 + threadIdx.x;
  if (e < E) atomicAdd(deg + ei[e], ew[e]);
}

__global__ void __launch_bounds__(256) dinv_kernel(float* __restrict__ deg, int N) {
  int i = blockIdx.x * 256 + threadIdx.x;
  if (i < N) {
    float d = deg[i];
    deg[i] = (d > 0.0f) ? rsqrtf(d) : 0.0f;  // in-place: deg becomes dinv
  }
}

__global__ void __launch_bounds__(256) weff_kernel(const int* __restrict__ ei,
                                                   const float* __restrict__ ew,
                                                   const float* __restrict__ dinv,
                                                   float* __restrict__ weff,
                                                   int E, int EN) {
  int e = blockIdx.x * 256 + threadIdx.x;
  if (e >= EN) return;
  if (e < E) {
    weff[e] = (1.0f - ALPHA) * dinv[ei[e]] * ew[e] * dinv[ei[E + e]];
  } else {
    float d = dinv[e - E];                   // self-loop edge for node e-E
    weff[e] = (1.0f - ALPHA) * d * d;
  }
}

// ---------------------------------------------------------------------------
// Pack a row-major [K x 64] weight matrix into WMMA B-fragment order:
//   Wp2[k2 * 64 + n] = (W[2*k2][n], W[2*k2+1][n])  as float2
// so each B fragment in the GEMM loop is a single global_load_b64.
// ---------------------------------------------------------------------------
__global__ void __launch_bounds__(256) pack_w_kernel(const float* __restrict__ W,
                                                     float* __restrict__ Wp,
                                                     int total /* (K/2)*64 */) {
  int i = blockIdx.x * 256 + threadIdx.x;
  if (i < total) {
    int k2 = i >> 6;          // /64
    int n  = i & 63;
    float2 v;
    v.x = W[(size_t)(2 * k2) * 64 + n];
    v.y = W[(size_t)(2 * k2 + 1) * 64 + n];
    ((float2*)Wp)[i] = v;
  }
}

// ---------------------------------------------------------------------------
// MLP:  h = relu(x @ W0 + b0) @ W1 + b1   via V_WMMA_F32_16X16X4_F32
// One wave32 per 16-row slab; 4 N-tiles of 16 cover H=64.
// A layout (32-bit 16x4): lanes 0-15 hold K=0(v0),K=1(v1); lanes 16-31 K=2,K=3.
// B layout (4x16, mirrors C/D half-split): v0 = K=0(lo)/K=2(hi), v1 = K=1/K=3.
// C/D layout: VGPR r = row (r + 8*half), column = (lane & 15) within tile.
// ---------------------------------------------------------------------------
#define WMMA_F32(A, B, C) \
  __builtin_amdgcn_wmma_f32_16x16x4_f32(false, (A), false, (B), (short)0, (C), false, false)

__global__ void __launch_bounds__(256) mlp_wmma_kernel(
    const float* __restrict__ x,
    const v2f* __restrict__ W0p,   // packed B fragments of W0, (F_IN/2)*64 float2
    const float* __restrict__ b0,
    const v2f* __restrict__ W1p,   // packed B fragments of W1, 32*64 float2
    const float* __restrict__ b1,
    float* __restrict__ h, int N, int F_IN) {
  __shared__ float stage[8 * 16 * 64];  // per-wave 16x64 h1 tile (32 KB)
  const int lane = threadIdx.x & 31;
  const int wave = threadIdx.x >> 5;
  const int half = lane >> 4;
  const int l16  = lane & 15;
  const int nslabs = (N + 15) >> 4;
  int slab = blockIdx.x * 8 + wave;
  if (slab >= nslabs) slab = nslabs - 1;   // clamp (redundant work) -> EXEC all-ones for WMMA
  const int m0 = slab * 16;
  int mrow = m0 + l16; if (mrow >= N) mrow = N - 1;

  // ---- GEMM1: [16 x F_IN] x [F_IN x 64] ----
  v8f a0 = {}, a1 = {}, a2 = {}, a3 = {};
  const float* xrow = x + (size_t)mrow * F_IN + 2 * half;
  for (int k = 0; k < F_IN; k += 4) {
    v2f a = *(const v2f*)(xrow + k);                         // global_load_b64
    const v2f* w = W0p + (size_t)((k >> 1) + half) * 64 + l16;
    a0 = WMMA_F32(a, w[0],  a0);                             // each B frag: 1x b64
    a1 = WMMA_F32(a, w[16], a1);
    a2 = WMMA_F32(a, w[32], a2);
    a3 = WMMA_F32(a, w[48], a3);
  }

  // ---- bias + ReLU, stage 16x64 tile row-major into LDS (C/D -> A transpose) ----
  float* sl = stage + wave * (16 * 64);
  {
    const float bb0 = b0[l16], bb1 = b0[16 + l16], bb2 = b0[32 + l16], bb3 = b0[48 + l16];
#pragma unroll
    for (int r = 0; r < 8; ++r) {
      int row = (r + 8 * half) * 64 + l16;
      float v;
      v = a0[r] + bb0; sl[row +  0] = v > 0.f ? v : 0.f;
      v = a1[r] + bb1; sl[row + 16] = v > 0.f ? v : 0.f;
      v = a2[r] + bb2; sl[row + 32] = v > 0.f ? v : 0.f;
      v = a3[r] + bb3; sl[row + 48] = v > 0.f ? v : 0.f;
    }
  }
  __syncthreads();

  // ---- GEMM2: [16 x 64] x [64 x 64], A-fragments from LDS ----
  v8f d0 = {}, d1 = {}, d2 = {}, d3 = {};
#pragma unroll 4
  for (int k = 0; k < 64; k += 4) {
    v2f a = *(const v2f*)(sl + l16 * 64 + k + 2 * half);     // ds_load_b64
    const v2f* w = W1p + (size_t)((k >> 1) + half) * 64 + l16;
    d0 = WMMA_F32(a, w[0],  d0);
    d1 = WMMA_F32(a, w[16], d1);
    d2 = WMMA_F32(a, w[32], d2);
    d3 = WMMA_F32(a, w[48], d3);
  }

  // ---- bias + store h ----
  const float c0 = b1[l16], c1 = b1[16 + l16], c2 = b1[32 + l16], c3 = b1[48 + l16];
#pragma unroll
  for (int r = 0; r < 8; ++r) {
    int row = m0 + r + 8 * half;
    if (row < N) {
      float* hp = h + (size_t)row * 64 + l16;
      hp[0]  = d0[r] + c0;
      hp[16] = d1[r] + c1;
      hp[32] = d2[r] + c2;
      hp[48] = d3[r] + c3;
    }
  }
}

// ---------------------------------------------------------------------------
// Propagation: z_out = ALPHA*h  then  z_out[col] += weff[e] * z_in[row]
// ---------------------------------------------------------------------------
__global__ void __launch_bounds__(256) scale_init_kernel(const float* __restrict__ h,
                                                         float* __restrict__ out, int n4) {
  int i = blockIdx.x * 256 + threadIdx.x;
  if (i < n4) {
    const float4 v = ((const float4*)h)[i];
    float4 o;
    o.x = ALPHA * v.x; o.y = ALPHA * v.y; o.z = ALPHA * v.z; o.w = ALPHA * v.w;
    ((float4*)out)[i] = o;
  }
}

__global__ void __launch_bounds__(256) prop_kernel(const int* __restrict__ ei,
                                                   const float* __restrict__ weff,
                                                   const float* __restrict__ zin,
                                                   float* __restrict__ zout,
                                                   int E, int EN) {
  const int e = blockIdx.x * 4 + (threadIdx.x >> 6);  // 4 edges per 256-thread block
  const int f = threadIdx.x & 63;                     // one feature per thread
  if (e >= EN) return;
  int r, c;
  if (e < E) { r = ei[e]; c = ei[E + e]; }
  else       { r = e - E; c = r; }                    // appended self loop
  const float v = weff[e] * zin[(size_t)r * 64 + f];  // L2-resident gather (z = 25.6 MB)
  atomicAdd(zout + (size_t)c * 64 + f, v);            // global_atomic_add_f32, no return
}

// ---------------------------------------------------------------------------
extern "C" void kernel_launch(void* const* d_in, const int* in_sizes, int n_in,
                              void* d_out, int out_size, void* d_ws, size_t ws_size,
                              hipStream_t stream) {
  const float* x  = (const float*)d_in[0];
  const int*   ei = (const int*)d_in[1];
  const float* ew = (const float*)d_in[2];
  const float* W0 = (const float*)d_in[3];
  const float* b0 = (const float*)d_in[4];
  const float* W1 = (const float*)d_in[5];
  const float* b1 = (const float*)d_in[6];
  (void)n_in; (void)out_size; (void)ws_size;

  const int H    = in_sizes[4];          // 64
  const int F_IN = in_sizes[3] / H;      // 512
  const int N    = in_sizes[0] / F_IN;   // 100000
  const int E    = in_sizes[1] / 2;      // 3200000
  const int EN   = E + N;
  (void)H;

  float* ws = (float*)d_ws;
  size_t off = 0;
  float* deg  = ws + off; off += ((size_t)N  + 3) & ~(size_t)3;  // -> becomes dinv
  float* weff = ws + off; off += ((size_t)EN + 3) & ~(size_t)3;
  float* h    = ws + off; off += (size_t)N * 64;
  float* buf0 = ws + off; off += (size_t)N * 64;
  float* W0p  = ws + off; off += (size_t)F_IN * 64;              // (F_IN/2)*64 float2
  float* W1p  = ws + off; off += (size_t)64 * 64;                // 32*64 float2
  float* out  = (float*)d_out;           // also used as the second z buffer

  dim3 blk(256);
  deg_init_kernel <<<(N  + 255) / 256, blk, 0, stream>>>(deg, N);
  deg_accum_kernel<<<(E  + 255) / 256, blk, 0, stream>>>(ei, ew, deg, E);
  dinv_kernel     <<<(N  + 255) / 256, blk, 0, stream>>>(deg, N);
  weff_kernel     <<<(EN + 255) / 256, blk, 0, stream>>>(ei, ew, deg, weff, E, EN);

  const int p0 = (F_IN / 2) * 64;        // W0 fragment pairs
  const int p1 = 32 * 64;                // W1 fragment pairs
  pack_w_kernel<<<(p0 + 255) / 256, blk, 0, stream>>>(W0, W0p, p0);
  pack_w_kernel<<<(p1 + 255) / 256, blk, 0, stream>>>(W1, W1p, p1);

  const int nslabs = (N + 15) / 16;
  mlp_wmma_kernel<<<(nslabs + 7) / 8, blk, 0, stream>>>(
      x, (const v2f*)W0p, b0, (const v2f*)W1p, b1, h, N, F_IN);

  const int n4 = (N * 64) / 4;
  for (int it = 0; it < NITER; ++it) {
    const float* zin = (it == 0) ? h : ((it & 1) ? buf0 : out);
    float* zout = (it & 1) ? out : buf0;           // it==9 (final) -> d_out
    scale_init_kernel<<<(n4 + 255) / 256, blk, 0, stream>>>(h, zout, n4);
    prop_kernel<<<(EN + 3) / 4, blk, 0, stream>>>(ei, weff, zin, zout, E, EN);
  }
}